// ModelNew_23656679867196
// MI455X (gfx1250) — compile-verified
//
#include <hip/hip_runtime.h>

typedef __attribute__((ext_vector_type(2))) float v2f;
typedef __attribute__((ext_vector_type(4))) float v4f;
typedef __attribute__((ext_vector_type(8))) float v8f;

#define CH 16           // input channels
#define OC 16           // output channels
#define HGT 256
#define WID 256
#define H_TILE 4
#define LDS_ROWS (H_TILE + 2)           // halo rows
#define LDS_W (WID + 2)                 // halo cols (zero pad left/right)
#define CSTRIDE 18                      // padded channel stride (floats): 72B entries,
                                        // 8B-aligned b64 loads, <=2-way bank conflicts
#define ENTRY(r, wc) (((r) * LDS_W + (wc)) * CSTRIDE)
#define LDS_FLOATS (LDS_ROWS * LDS_W * CSTRIDE)   // 27864 floats = 111456 B

// Implicit GEMM: D(16oc x 16px) = A(16oc x 144) * B(144 x 16px) + bias - 0.7,
// K = (channel, tap), consumed as 36 x V_WMMA_F32_16X16X4_F32 per tile.
__global__ __launch_bounds__(256)
void conv3x3_bias_sub_mish_wmma(const float* __restrict__ x,
                                const float* __restrict__ wgt,
                                const float* __restrict__ bias,
                                float* __restrict__ out) {
  extern __shared__ float xs[];   // [LDS_ROWS][LDS_W][CSTRIDE]

  const int blk = blockIdx.x;
  const int hb  = blk & 63;          // H / H_TILE = 64
  const int n   = blk >> 6;
  const int h0  = hb * H_TILE;
  const int tid = threadIdx.x;

  // ---- zero-fill LDS (provides all halo zeros), vectorized b128 ----
  v4f* xs4 = (v4f*)xs;
  const v4f z4 = {0.0f, 0.0f, 0.0f, 0.0f};
  for (int e = tid; e < LDS_FLOATS / 4; e += 256) xs4[e] = z4;

  const int lane = tid & 31;
  const int half = lane >> 4;   // fragment k_local base = 2*half
  const int lrow = lane & 15;   // A: row (oc); B/D: column (pixel)

  // ---- A fragments (weights) in VGPRs for the whole kernel ----
  // chunk j: tap t = j>>2, channel base c0 = (j&3)*4 + 2*half
  v2f a_frag[36];
#pragma unroll
  for (int j = 0; j < 36; ++j) {
    const int t  = j >> 2;
    const int c0 = (j & 3) * 4 + 2 * half;
    // conv_weight (O, I, 3, 3): offset (o*16 + c)*9 + t
    a_frag[j].x = wgt[(lrow * CH + c0)     * 9 + t];
    a_frag[j].y = wgt[(lrow * CH + c0 + 1) * 9 + t];
  }

  // accumulator seed: bias[oc] - (0.5 + 0.2); D VGPR i holds oc = i + 8*half
  float binit[8];
#pragma unroll
  for (int i = 0; i < 8; ++i) binit[i] = bias[i + 8 * half] - 0.7f;

  __syncthreads();   // memset visible before interior copy

  // ---- stage interior: float4 global loads, scatter into [row][w][c] ----
  // 96 (c,row) pairs x 64 float4 chunks = 6144 units
  for (int e = tid; e < CH * LDS_ROWS * (WID / 4); e += 256) {
    const int k  = e & 63;        // float4 index along w
    const int pr = e >> 6;
    const int c  = pr & 15;
    const int r  = pr >> 4;       // 0..5
    const int gh = h0 + r - 1;
    if (gh >= 0 && gh < HGT) {
      const v4f* row = (const v4f*)(x + (((n * CH + c) * HGT) + gh) * WID);
      const v4f v = row[k];
      float* dst = &xs[ENTRY(r, 4 * k + 1) + c];
      dst[0 * CSTRIDE] = v.x;
      dst[1 * CSTRIDE] = v.y;
      dst[2 * CSTRIDE] = v.z;
      dst[3 * CSTRIDE] = v.w;
    }
  }

  __syncthreads();

  const int wv    = tid >> 5;     // wave 0..7
  const int r_out = wv >> 1;      // output row within block tile: 0..3
  const int tile0 = (wv & 1) * 8; // w-tile range start

  for (int tt = 0; tt < 8; ++tt) {
    const int w0 = (tile0 + tt) * 16;

    v8f acc;
#pragma unroll
    for (int i = 0; i < 8; ++i) acc[i] = binit[i];

#pragma unroll
    for (int t = 0; t < 9; ++t) {
      const int dy = t / 3, dx = t % 3;
      // one aligned b64 per chunk: channels (c0+2*half, c0+2*half+1) adjacent
      const float* src = &xs[ENTRY(r_out + dy, w0 + lrow + dx) + 2 * half];
#pragma unroll
      for (int cc = 0; cc < 4; ++cc) {
        const v2f b = *(const v2f*)(src + cc * 4);
        acc = __builtin_amdgcn_wmma_f32_16x16x4_f32(
            false, a_frag[t * 4 + cc], false, b, (short)0, acc, false, false);
      }
    }

    // ---- epilogue: Mish via u = e^y (e^y + 2): y * u / (u + 2) ----
    const int h  = h0 + r_out;
    const int gw = w0 + lrow;
    float* op = out + (((n * OC + 8 * half) * HGT) + h) * WID + gw;
#pragma unroll
    for (int i = 0; i < 8; ++i) {
      const float y  = acc[i];
      const float et = __expf(y);
      const float u  = et * (et + 2.0f);
      const float m  = y * __fdividef(u, u + 2.0f);
      op[i * (HGT * WID)] = (y >= 30.0f) ? y : m;   // exact (and overflow-safe) tail
    }
  }
}

extern "C" void kernel_launch(void* const* d_in, const int* in_sizes, int n_in,
                              void* d_out, int out_size, void* d_ws, size_t ws_size,
                              hipStream_t stream) {
  const float* x    = (const float*)d_in[0];   // (32,16,256,256)
  const float* wgt  = (const float*)d_in[1];   // (16,16,3,3)
  const float* bias = (const float*)d_in[2];   // (16,)
  float* out = (float*)d_out;

  const int nblocks = 32 * (HGT / H_TILE);          // 2048
  const size_t lds_bytes = LDS_FLOATS * sizeof(float);  // ~109 KB dynamic LDS
  conv3x3_bias_sub_mish_wmma<<<nblocks, 256, lds_bytes, stream>>>(x, wgt, bias, out);
}